// Filter_52888227282958
// MI455X (gfx1250) — compile-verified
//
#include <hip/hip_runtime.h>
#include <hip/hip_bf16.h>
#include <stdint.h>

// ---------------------------------------------------------------------------
// KPN per-pixel filtering, MI455X (gfx1250).
// Memory-bound (AI ~2.2 FLOP/B): target = one HBM touch per byte.
//  - k[19] per pixel cached in VGPRs (reused across 32 channels)
//  - x halo tiles staged LDS via gfx1250 async-to-LDS, double buffered
//  - skip/out streamed non-temporally
// ---------------------------------------------------------------------------

#define NCH    64
#define NFILT  2
#define BATCH  4
#define HH     256
#define WW     256
#define HWSZ   (HH * WW)          // 65536
#define KSLOTS 19

#define TH 16                     // tile rows
#define TW 32                     // tile cols
#define HR 18                     // halo rows  (TH + 2)
#define HC 34                     // halo cols  (TW + 2)
#define HSTRIDE 36                // padded LDS row stride (floats)
#define FSTRIDE (HR * HSTRIDE)    // 648 floats per filter tile
#define BUFSZ   (NFILT * FSTRIDE) // 1296 floats per buffer
#define NHALO   (NFILT * HR * HC) // 1224 async elements per channel
#define CSPLIT  2                 // channel split across blocks

typedef float v2f __attribute__((ext_vector_type(2)));
typedef __attribute__((address_space(1))) int* gint1_t;  // global int*
typedef __attribute__((address_space(3))) int* lint3_t;  // LDS int*

// gfx1250 async copy global->LDS (ASYNCcnt path). Builtin if the toolchain
// declares it, else raw CDNA5 asm per CDNA5 ISA.
__device__ __forceinline__ void async_ld_f32(const float* gsrc, float* ldst) {
#if __has_builtin(__builtin_amdgcn_global_load_async_to_lds_b32)
  __builtin_amdgcn_global_load_async_to_lds_b32(
      (gint1_t)(uintptr_t)gsrc, (lint3_t)(uintptr_t)ldst, 0, 0);
#else
  // flat LDS address low 32 bits == LDS byte offset (ISA 10.2 aperture map)
  unsigned loff = (unsigned)(uintptr_t)ldst;
  asm volatile("global_load_async_to_lds_b32 %0, %1, off"
               :: "v"(loff), "v"((unsigned long long)(uintptr_t)gsrc)
               : "memory");
#endif
}

__device__ __forceinline__ void wait_async0() {
#if __has_builtin(__builtin_amdgcn_s_wait_asynccnt)
  __builtin_amdgcn_s_wait_asynccnt(0);
#else
  asm volatile("s_wait_asynccnt 0" ::: "memory");
#endif
}

__global__ __launch_bounds__(256)
void kpn_filter_kernel(const float* __restrict__ kk,
                       const float* __restrict__ xx,
                       const float* __restrict__ skip,
                       float* __restrict__ out) {
  __shared__ float xs[2 * BUFSZ];

  const int tid = threadIdx.x;
  const int bz  = blockIdx.z;
  const int b   = bz >> 1;
  const int c0  = (bz & 1) * (NCH / CSPLIT);
  const int h0  = blockIdx.y * TH;
  const int w0  = blockIdx.x * TW;

  const int r    = tid >> 4;          // 0..15 tile row
  const int c2   = (tid & 15) << 1;   // 0..30 even tile col (pixel pair)
  const int gpix = (h0 + r) * WW + (w0 + c2);

  // Zero LDS once: covers padded cols + out-of-image halo slots; async loads
  // only ever write in-range slots, so the zeros persist (correct padding).
  for (int i = tid; i < 2 * BUFSZ; i += 256) xs[i] = 0.0f;

  // Precompute this thread's valid halo-load slots (<=5 per thread).
  int     nv = 0;
  int     ldso[5];
  int64_t gofs[5];
#pragma unroll
  for (int t = 0; t < 5; ++t) {
    int idx = tid + t * 256;
    if (idx < NHALO) {
      int f  = (idx >= HR * HC) ? 1 : 0;
      int e  = idx - f * (HR * HC);
      int hr = e / HC;
      int hc = e - hr * HC;
      int gh = h0 - 1 + hr;
      int gw = w0 - 1 + hc;
      if ((unsigned)gh < (unsigned)HH && (unsigned)gw < (unsigned)WW) {
        ldso[nv] = f * FSTRIDE + hr * HSTRIDE + hc;
        gofs[nv] = (int64_t)f * (NCH * HWSZ) + (int64_t)gh * WW + gw;
        ++nv;
      }
    }
  }

  const float* xb = xx   + (int64_t)b * (NFILT * NCH) * HWSZ;
  const float* kb = kk   + (int64_t)b * KSLOTS * HWSZ + gpix;
  const float* sb = skip + (int64_t)b * NCH * HWSZ    + gpix;
  float*       ob = out  + (int64_t)b * NCH * HWSZ    + gpix;

  // Per-pixel predicted kernel -> VGPRs, reused for all channels.
  v2f kv[KSLOTS];
#pragma unroll
  for (int s = 0; s < KSLOTS; ++s)
    kv[s] = __builtin_nontemporal_load((const v2f*)(kb + s * HWSZ));

  // Prologue: stage first channel's halo tiles into buffer 0.
  {
    const float* src = xb + (int64_t)c0 * HWSZ;
#pragma unroll
    for (int i = 0; i < 5; ++i)
      if (i < nv) async_ld_f32(src + gofs[i], &xs[ldso[i]]);
  }
  wait_async0();
  __syncthreads();

#pragma unroll 1
  for (int cc = 0; cc < NCH / CSPLIT; ++cc) {
    const int c  = c0 + cc;
    const int p  = cc & 1;
    const int np = p ^ 1;

    // Prefetch next channel into the other LDS buffer (overlaps compute).
    if (cc + 1 < NCH / CSPLIT) {
      const float* src = xb + (int64_t)(c + 1) * HWSZ;
#pragma unroll
      for (int i = 0; i < 5; ++i)
        if (i < nv) async_ld_f32(src + gofs[i], &xs[np * BUFSZ + ldso[i]]);
    }

    v2f sk = __builtin_nontemporal_load((const v2f*)(sb + (int64_t)c * HWSZ));
    float a0 = kv[18].x * sk.x;
    float a1 = kv[18].y * sk.y;

    const float* base = &xs[p * BUFSZ + r * HSTRIDE + c2];
#pragma unroll
    for (int f = 0; f < NFILT; ++f) {
#pragma unroll
      for (int di = 0; di < 3; ++di) {
        const float* row = base + f * FSTRIDE + di * HSTRIDE;
        v2f t01 = *(const v2f*)(row);       // cols c2, c2+1   (ds_load_b64)
        v2f t23 = *(const v2f*)(row + 2);   // cols c2+2, c2+3 (ds_load_b64)
        const int s = f * 9 + di * 3;
        a0 += kv[s].x * t01.x + kv[s + 1].x * t01.y + kv[s + 2].x * t23.x;
        a1 += kv[s].y * t01.y + kv[s + 1].y * t23.x + kv[s + 2].y * t23.y;
      }
    }

    v2f res; res.x = a0; res.y = a1;
    __builtin_nontemporal_store(res, (v2f*)(ob + (int64_t)c * HWSZ));

    // Next-channel asyncs must land, and all waves must finish reading buf p
    // before it is overwritten by the next iteration's prefetch.
    wait_async0();
    __syncthreads();
  }
}

extern "C" void kernel_launch(void* const* d_in, const int* in_sizes, int n_in,
                              void* d_out, int out_size, void* d_ws, size_t ws_size,
                              hipStream_t stream) {
  const float* k    = (const float*)d_in[0];  // [B, 19, H, W]
  const float* x    = (const float*)d_in[1];  // [B, 128, H, W]
  const float* skip = (const float*)d_in[2];  // [B, 64, H, W]
  float*       out  = (float*)d_out;          // [B, 64, H, W]

  dim3 grid(WW / TW, HH / TH, BATCH * CSPLIT);
  kpn_filter_kernel<<<grid, 256, 0, stream>>>(k, x, skip, out);
}